// CapsFC_48455821033782
// MI455X (gfx1250) — compile-verified
//
#include <hip/hip_runtime.h>
#include <hip/hip_bf16.h>

// Problem constants (from reference setup_inputs)
#define BATCH   256
#define IN_CAP  1152
#define IN_LEN  8
#define OUT_CAP 10
#define OUT_LEN 16
#define MDIM    (OUT_CAP * OUT_LEN)   // 160 rows of the per-i GEMM
#define ITERS   4

typedef __attribute__((ext_vector_type(2))) float v2f;
typedef __attribute__((ext_vector_type(8))) float v8f;
typedef __attribute__((ext_vector_type(4))) unsigned int v4u;
typedef __attribute__((ext_vector_type(8))) int v8i_t;
typedef __attribute__((ext_vector_type(4))) int v4i_t;

#if defined(__has_builtin)
#if __has_builtin(__builtin_amdgcn_tensor_load_to_lds) && \
    __has_builtin(__builtin_amdgcn_s_wait_tensorcnt)
#define CAPS_USE_TDM 1
#endif
#endif

// ---------------------------------------------------------------------------
// Stage 1: u_hat[b][i][o*16+k] = sum_l W[i][o][k][l] * x[b][i][l]
// Batched over i: 1152 GEMMs of (160 x 8) * (8 x 256), done with
// V_WMMA_F32_16X16X4_F32 tiles (M=16 over (o,k), N=16 over batch, K=4 x2).
// One wave per 16x16 output tile; 8 waves (256 threads) per block.
// Result is transposed through LDS so stores are 64B-contiguous in the
// routing-friendly [b][i][m] layout.
// ---------------------------------------------------------------------------
__global__ __launch_bounds__(256)
void caps_uhat_wmma(const float* __restrict__ x,
                    const float* __restrict__ W,
                    float* __restrict__ uhat) {
    __shared__ float tile[8 * 16 * 20];  // 8 waves * 16 rows(b) * pitch 20 (m)

    const int wib  = threadIdx.x >> 5;           // wave in block 0..7
    const int lane = threadIdx.x & 31;
    const int half = lane >> 4;                  // 0: lanes 0-15, 1: lanes 16-31
    const int ln   = lane & 15;

    // total wave-tiles = IN_CAP * (MDIM/16) * (BATCH/16) = 1152*10*16 = 184320
    const int wid = blockIdx.x * 8 + wib;
    const int bt  = wid % (BATCH / 16);          // batch tile 0..15
    const int mt  = (wid / (BATCH / 16)) % (MDIM / 16); // m tile 0..9
    const int i   = wid / ((BATCH / 16) * (MDIM / 16)); // capsule 0..1151

    // ---- A operand: W rows. A is 16x4 (MxK). lanes 0-15 hold K={0,1},
    //      lanes 16-31 hold K={2,3} (VGPR0=Keven, VGPR1=Kodd).
    const int m = mt * 16 + ln;                  // global row (o*16+k)
    const float* wrow = W + ((size_t)i * MDIM + m) * IN_LEN;
    const float2 af0 = *reinterpret_cast<const float2*>(wrow + 2 * half);      // K-step 0
    const float2 af1 = *reinterpret_cast<const float2*>(wrow + 4 + 2 * half);  // K-step 1

    // ---- B operand: x columns. B is 4x16 (KxN). lanes 0-15: K={0,1},
    //      lanes 16-31: K={2,3}; N = ln.
    const int bcol = bt * 16 + ln;
    const float* xrow = x + ((size_t)bcol * IN_CAP + i) * IN_LEN;
    const float2 bf0 = *reinterpret_cast<const float2*>(xrow + 2 * half);
    const float2 bf1 = *reinterpret_cast<const float2*>(xrow + 4 + 2 * half);

    v2f a0 = {af0.x, af0.y};
    v2f a1 = {af1.x, af1.y};
    v2f b0 = {bf0.x, bf0.y};
    v2f b1 = {bf1.x, bf1.y};

    v8f acc = {};
    // D = A*B + C, fp32 accumulate, K=8 in two K=4 steps
    acc = __builtin_amdgcn_wmma_f32_16x16x4_f32(false, a0, false, b0,
                                                (short)0, acc, false, false);
    acc = __builtin_amdgcn_wmma_f32_16x16x4_f32(false, a1, false, b1,
                                                (short)0, acc, false, false);

    // ---- transpose through LDS: D VGPR j holds (m_local = j + 8*half, n = ln)
    float* slab = tile + wib * (16 * 20);
#pragma unroll
    for (int j = 0; j < 8; ++j) {
        slab[ln * 20 + (j + 8 * half)] = acc[j];
    }
    __syncthreads();

    // ---- store: each lane writes one 16B segment pair; lane pair (2r,2r+1)
    //      covers the 64B row for local batch r.
    const int r  = lane >> 1;                    // local batch row 0..15
    const int s0 = (lane & 1) * 2;               // segment 0/2
    const int bg = bt * 16 + r;
    float* orow = uhat + ((size_t)bg * IN_CAP + i) * MDIM + mt * 16;
    *reinterpret_cast<float4*>(orow + s0 * 4) =
        *reinterpret_cast<const float4*>(slab + r * 20 + s0 * 4);
    *reinterpret_cast<float4*>(orow + (s0 + 1) * 4) =
        *reinterpret_cast<const float4*>(slab + r * 20 + (s0 + 1) * 4);
}

// ---------------------------------------------------------------------------
// Stage 2: dynamic routing. One wave per (b,o) pair (2560 independent
// problems). 4 waves per block, each wave stages its 73.7 KB u_hat slice
// u_hat[b, :, o, :] into LDS once (4 * 73728 B = 294912 B <= 320 KB/WGP),
// then runs all 4 routing iterations out of LDS + registers.
// The slice is a 2D tile (16 x 1152 elements, row stride 160 elements),
// moved by a single Tensor Data Mover descriptor when available.
// b_ij column lives in registers: lane holds i = lane + 32*t, t = 0..35.
// ---------------------------------------------------------------------------
__global__ __launch_bounds__(128)
void caps_route(const float* __restrict__ uhat,
                float* __restrict__ out) {
    extern __shared__ float smem[];              // 4 * IN_CAP * OUT_LEN floats

    const int wib  = threadIdx.x >> 5;           // 0..3
    const int lane = threadIdx.x & 31;
    const int g    = blockIdx.x * 4 + wib;       // 0..2559
    const int b    = g / OUT_CAP;
    const int o    = g % OUT_CAP;

    float* slab = smem + (size_t)wib * (IN_CAP * OUT_LEN);
    const float* src = uhat + ((size_t)b * IN_CAP) * MDIM + o * OUT_LEN;

#if defined(CAPS_USE_TDM)
    // ---- TDM: one descriptor moves the whole 2D tile Global -> LDS.
    //      D# group0: count=1, lds_addr, global_addr[56:0], type=2 (image).
    //      D# group1: data_size=2 (4B), tensor_dim0=16, tensor_dim1=1152,
    //                 tile_dim0=16, tile_dim1=1152, tensor_dim0_stride=160.
    {
        const unsigned long long gaddr = (unsigned long long)(uintptr_t)src;
        const unsigned int       laddr = (unsigned int)(uintptr_t)slab;

        v4u g0 = {1u,                                   // count=1, flags=0
                  laddr,                                // lds_addr (bytes)
                  (unsigned int)(gaddr & 0xFFFFFFFFu),  // global_addr[31:0]
                  (unsigned int)((gaddr >> 32) & 0x1FFFFFFu) | (2u << 30)};

        v8i_t g1 = {(int)(2u << 16),          // data_size=4B; mask/flags/pad=0
                    (int)(16u << 16),         // tensor_dim0[15:0] in [63:48]
                    (int)(1152u << 16),       // tensor_dim0[31:16]=0, dim1[15:0]
                    (int)(16u << 16),         // dim1[31:16]=0, tile_dim0=16
                    (int)1152,                // tile_dim1=1152, tile_dim2=0
                    (int)160,                 // tensor_dim0_stride[31:0]=160
                    0, 0};                    // stride hi, dim1_stride=0

        v4i_t gz  = {0, 0, 0, 0};             // groups 2/3 unused (2D tensor)
        v8i_t gz8 = {0, 0, 0, 0, 0, 0, 0, 0};

#if defined(__clang_major__) && (__clang_major__ >= 23)
        // upstream clang-23 / therock-10.0 HIP headers: 6-arg form
        __builtin_amdgcn_tensor_load_to_lds(g0, g1, gz, gz, gz8, 0);
#else
        // ROCm 7.2 clang-22: 5-arg form
        __builtin_amdgcn_tensor_load_to_lds(g0, g1, gz, gz, 0);
#endif
        __builtin_amdgcn_s_wait_tensorcnt(0);
    }
#else
    // ---- fallback: staged copy through VGPRs (float4 granularity).
    {
        for (int idx = lane; idx < IN_CAP * 4; idx += 32) {
            const int ii  = idx >> 2;
            const int seg = idx & 3;
            float4 v = *reinterpret_cast<const float4*>(src + (size_t)ii * MDIM + seg * 4);
            *reinterpret_cast<float4*>(slab + ii * OUT_LEN + seg * 4) = v;
        }
        __syncthreads();
    }
#endif

    // ---- routing state: b_ij in registers, i = lane + 32*t
    float bij[36];
#pragma unroll
    for (int t = 0; t < 36; ++t) bij[t] = 0.0f;

    float v0[OUT_LEN];                           // routed output (all lanes)

    for (int it = 0; it < ITERS; ++it) {
        // softmax max over i (numerical stability)
        float mx = -3.0e38f;
#pragma unroll
        for (int t = 0; t < 36; ++t) mx = fmaxf(mx, bij[t]);
#pragma unroll
        for (int off = 16; off >= 1; off >>= 1)
            mx = fmaxf(mx, __shfl_xor(mx, off, 32));

        // single pass: Z = sum exp, sk[k] = sum exp * u
        float Zp = 0.0f;
        float sk[OUT_LEN];
#pragma unroll
        for (int k = 0; k < OUT_LEN; ++k) sk[k] = 0.0f;

#pragma unroll 4
        for (int t = 0; t < 36; ++t) {
            const float e = __expf(bij[t] - mx);
            Zp += e;
            const float* urow = slab + (lane + 32 * t) * OUT_LEN;
            float4 u0 = *reinterpret_cast<const float4*>(urow + 0);
            float4 u1 = *reinterpret_cast<const float4*>(urow + 4);
            float4 u2 = *reinterpret_cast<const float4*>(urow + 8);
            float4 u3 = *reinterpret_cast<const float4*>(urow + 12);
            sk[0]  = fmaf(e, u0.x, sk[0]);  sk[1]  = fmaf(e, u0.y, sk[1]);
            sk[2]  = fmaf(e, u0.z, sk[2]);  sk[3]  = fmaf(e, u0.w, sk[3]);
            sk[4]  = fmaf(e, u1.x, sk[4]);  sk[5]  = fmaf(e, u1.y, sk[5]);
            sk[6]  = fmaf(e, u1.z, sk[6]);  sk[7]  = fmaf(e, u1.w, sk[7]);
            sk[8]  = fmaf(e, u2.x, sk[8]);  sk[9]  = fmaf(e, u2.y, sk[9]);
            sk[10] = fmaf(e, u2.z, sk[10]); sk[11] = fmaf(e, u2.w, sk[11]);
            sk[12] = fmaf(e, u3.x, sk[12]); sk[13] = fmaf(e, u3.y, sk[13]);
            sk[14] = fmaf(e, u3.z, sk[14]); sk[15] = fmaf(e, u3.w, sk[15]);
        }

        // wave32 tree-reduce Z and the 16 partial sums (all lanes get result)
#pragma unroll
        for (int off = 16; off >= 1; off >>= 1) {
            Zp += __shfl_xor(Zp, off, 32);
#pragma unroll
            for (int k = 0; k < OUT_LEN; ++k)
                sk[k] += __shfl_xor(sk[k], off, 32);
        }

        // s_j = sk / Z ; v = squash(s_j)
        const float invZ = __frcp_rn(Zp);
        float sq = 0.0f;
#pragma unroll
        for (int k = 0; k < OUT_LEN; ++k) {
            sk[k] *= invZ;
            sq = fmaf(sk[k], sk[k], sq);
        }
        const float scale = (sq / (1.0f + sq)) * __frsqrt_rn(sq + 1e-9f);
#pragma unroll
        for (int k = 0; k < OUT_LEN; ++k) v0[k] = sk[k] * scale;

        // agreement update (dead on last iteration in the reference)
        if (it < ITERS - 1) {
#pragma unroll 4
            for (int t = 0; t < 36; ++t) {
                const float* urow = slab + (lane + 32 * t) * OUT_LEN;
                float4 u0 = *reinterpret_cast<const float4*>(urow + 0);
                float4 u1 = *reinterpret_cast<const float4*>(urow + 4);
                float4 u2 = *reinterpret_cast<const float4*>(urow + 8);
                float4 u3 = *reinterpret_cast<const float4*>(urow + 12);
                float d = 0.0f;
                d = fmaf(u0.x, v0[0], d);  d = fmaf(u0.y, v0[1], d);
                d = fmaf(u0.z, v0[2], d);  d = fmaf(u0.w, v0[3], d);
                d = fmaf(u1.x, v0[4], d);  d = fmaf(u1.y, v0[5], d);
                d = fmaf(u1.z, v0[6], d);  d = fmaf(u1.w, v0[7], d);
                d = fmaf(u2.x, v0[8], d);  d = fmaf(u2.y, v0[9], d);
                d = fmaf(u2.z, v0[10], d); d = fmaf(u2.w, v0[11], d);
                d = fmaf(u3.x, v0[12], d); d = fmaf(u3.y, v0[13], d);
                d = fmaf(u3.z, v0[14], d); d = fmaf(u3.w, v0[15], d);
                bij[t] += d;
            }
        }
    }

    // ---- write v[b][o][0:16] (lane 0 of each wave)
    if (lane == 0) {
        float* dst = out + ((size_t)b * OUT_CAP + o) * OUT_LEN;
#pragma unroll
        for (int k = 0; k < OUT_LEN; k += 4) {
            float4 w4 = {v0[k], v0[k + 1], v0[k + 2], v0[k + 3]};
            *reinterpret_cast<float4*>(dst + k) = w4;
        }
    }
}

// ---------------------------------------------------------------------------
extern "C" void kernel_launch(void* const* d_in, const int* in_sizes, int n_in,
                              void* d_out, int out_size, void* d_ws, size_t ws_size,
                              hipStream_t stream) {
    const float* x = (const float*)d_in[0];   // (256, 1152, 8)
    const float* W = (const float*)d_in[1];   // (1152, 10, 16, 8)
    float* out  = (float*)d_out;              // (256, 10, 16)
    float* uhat = (float*)d_ws;               // (256, 1152, 160) = 188.7 MB

    // Stage 1: 184320 wave-tiles / 8 waves per block = 23040 blocks
    caps_uhat_wmma<<<23040, 256, 0, stream>>>(x, W, uhat);

    // Stage 2: 2560 (b,o) waves / 4 per block = 640 blocks,
    // 4 * 1152 * 16 * 4 B = 294912 B dynamic LDS per block
    caps_route<<<640, 128, 4 * IN_CAP * OUT_LEN * sizeof(float), stream>>>(uhat, out);
}